// RelativePositionEncoder_69999376990394
// MI455X (gfx1250) — compile-verified
//
#include <hip/hip_runtime.h>
#include <hip/hip_bf16.h>

// Problem constants (from reference)
#define B_DIM   2
#define L_DIM   768
#define R_MAX   32
#define S_MAX   2
#define TOKEN_Z 128
#define IN_F    139   // 66 rel_pos | 66 rel_token | 1 same_entity | 6 rel_chain

typedef float float4v __attribute__((ext_vector_type(4)));
typedef int   v4i     __attribute__((vector_size(16)));          // matches builtin param type
typedef __attribute__((address_space(1))) v4i* gptr_v4i;         // global (prints as __device__)
typedef __attribute__((address_space(3))) v4i* lptr_v4i;         // LDS

__device__ __forceinline__ int iclamp(int v, int lo, int hi) {
    return v < lo ? lo : (v > hi ? hi : v);
}

__global__ __launch_bounds__(256) void relpos_encoder_kernel(
    const int* __restrict__ asym_id,
    const int* __restrict__ residue_index,
    const int* __restrict__ entity_id,
    const int* __restrict__ cyclic_period,
    const int* __restrict__ token_index,
    const int* __restrict__ sym_id,
    const float* __restrict__ W,        // [139, 128]
    float* __restrict__ out,            // [B, L, L, 128]
    int nGroups)                        // B*L*L / 8
{
    __shared__ float Wlds[IN_F * TOKEN_Z];   // 71,168 bytes

    const int tid = threadIdx.x;

    // ---- Stage W into LDS via CDNA5 async global->LDS DMA (ASYNCcnt path) ----
    const int nVec = (IN_F * TOKEN_Z) / 4;   // 4448 float4's
#if __has_builtin(__builtin_amdgcn_global_load_async_to_lds_b128)
    for (int v = tid; v < nVec; v += 256) {
        __builtin_amdgcn_global_load_async_to_lds_b128(
            (gptr_v4i)(W + 4 * v),
            (lptr_v4i)(Wlds + 4 * v),
            /*offset=*/0, /*cpol=*/0);
    }
  #if __has_builtin(__builtin_amdgcn_s_wait_asynccnt)
    __builtin_amdgcn_s_wait_asynccnt(0);
  #else
    asm volatile("s_wait_asynccnt 0" ::: "memory");
  #endif
#else
    for (int v = tid; v < nVec; v += 256) {
        ((float4v*)Wlds)[v] = ((const float4v*)W)[v];
    }
#endif
    __syncthreads();

    const int lane      = tid & 31;   // z chunk: 4 floats per lane
    const int waveInBlk = tid >> 5;   // 8 pairs per block-iteration

    for (int grp = blockIdx.x; grp < nGroups; grp += gridDim.x) {
        const long long pair = (long long)grp * 8 + waveInBlk;   // < B*L*L
        const int b   = (int)(pair / (L_DIM * L_DIM));
        const int rem = (int)(pair % (L_DIM * L_DIM));
        const int i   = rem / L_DIM;
        const int j   = rem - i * L_DIM;

        const int bi = b * L_DIM + i;
        const int bj = b * L_DIM + j;

        const int ai = asym_id[bi],       aj = asym_id[bj];
        const int ri = residue_index[bi], rj = residue_index[bj];
        const int ei = entity_id[bi],     ej = entity_id[bj];
        const int ti = token_index[bi],   tj = token_index[bj];
        const int si = sym_id[bi],        sj = sym_id[bj];
        const int cp = cyclic_period[bj];   // period broadcast is [:, None, :] -> indexed by j

        const bool same_chain = (ai == aj);
        const bool same_res   = (ri == rj);
        const float se        = (ei == ej) ? 1.0f : 0.0f;

        // cyclic wrap: rel -= period * round_half_even(rel / period)
        const int period = (cp > 0) ? cp : 10000;
        int rel = ri - rj;
        rel -= period * (int)rintf((float)rel / (float)period);

        const int d_res = iclamp(rel + R_MAX, 0, 2 * R_MAX);
        const int r0    = same_chain ? d_res : (2 * R_MAX + 1);                      // [0,65]

        const int d_tok = iclamp(ti - tj + R_MAX, 0, 2 * R_MAX);
        const int r1    = 66 + ((same_chain && same_res) ? d_tok : (2 * R_MAX + 1)); // [66,131]

        const int d_ch  = iclamp(si - sj + S_MAX, 0, 2 * S_MAX);
        const int r2    = 133 + (same_chain ? (2 * S_MAX + 1) : d_ch);               // [133,138]

        // Gather 4 rows from LDS (512B rows -> conflict-free at lane*16B)
        const float4v a = ((const float4v*)(Wlds + r0  * TOKEN_Z))[lane];
        const float4v c = ((const float4v*)(Wlds + r1  * TOKEN_Z))[lane];
        const float4v e = ((const float4v*)(Wlds + 132 * TOKEN_Z))[lane];
        const float4v d = ((const float4v*)(Wlds + r2  * TOKEN_Z))[lane];

        float4v s;
        s.x = fmaf(se, e.x, a.x + c.x + d.x);
        s.y = fmaf(se, e.y, a.y + c.y + d.y);
        s.z = fmaf(se, e.z, a.z + c.z + d.z);
        s.w = fmaf(se, e.w, a.w + c.w + d.w);

        ((float4v*)out)[pair * (TOKEN_Z / 4) + lane] = s;   // global_store_b128
    }
}

extern "C" void kernel_launch(void* const* d_in, const int* in_sizes, int n_in,
                              void* d_out, int out_size, void* d_ws, size_t ws_size,
                              hipStream_t stream) {
    (void)in_sizes; (void)n_in; (void)out_size; (void)d_ws; (void)ws_size;

    const int* asym_id       = (const int*)d_in[0];
    const int* residue_index = (const int*)d_in[1];
    const int* entity_id     = (const int*)d_in[2];
    const int* cyclic_period = (const int*)d_in[3];
    const int* token_index   = (const int*)d_in[4];
    const int* sym_id        = (const int*)d_in[5];
    const float* W           = (const float*)d_in[6];
    float* out               = (float*)d_out;

    const int nPairs  = B_DIM * L_DIM * L_DIM;   // 1,179,648
    const int nGroups = nPairs / 8;              // 147,456 (8 pairs / block-iter)
    const int blocks  = 9216;                    // 16 iterations per block -> W stage amortized

    relpos_encoder_kernel<<<blocks, 256, 0, stream>>>(
        asym_id, residue_index, entity_id, cyclic_period, token_index, sym_id,
        W, out, nGroups);
}